// UCBNorm_24558622998529
// MI455X (gfx1250) — compile-verified
//
#include <hip/hip_runtime.h>
#include <stdint.h>

// UCBNorm for MI455X (gfx1250).
// B=16, T=512, D=256, K=8. Output [B,T,D] f32.
// No matmul structure -> WMMA inapplicable; the CDNA5 win is data movement:
// the Tensor Data Mover DMAs 64x256 f32 tiles of x into LDS (one
// tensor_load_to_lds per tile, overlapped with parameter setup), compute reads
// from LDS, and all reductions are fixed-order (deterministic, no atomics).
// x (8 MB) stays L2-resident across the 3 sweeps (192 MB L2), so the kernels
// are transcendental-throughput bound, not HBM bound.

#define UCB_B 16
#define UCB_T 512
#define UCB_D 256
#define UCB_K 8
#define UCB_ROWS 64                    // tile rows (t-direction)
#define UCB_NB   (UCB_B * UCB_T / UCB_ROWS)   // 128 tiles over all (b,t)
#define UCB_EPS  1e-3f
#define UCB_CNORM 0.99950037468787f    // 1/sqrt(1+EPS)

#if __has_builtin(__builtin_amdgcn_tensor_load_to_lds) && __has_builtin(__builtin_amdgcn_s_wait_tensorcnt)
#define UCB_USE_TDM 1
#endif

typedef unsigned int v4u __attribute__((ext_vector_type(4)));
typedef int          v8i __attribute__((ext_vector_type(8)));
typedef int          v4i __attribute__((ext_vector_type(4)));

#ifdef UCB_USE_TDM
// Build a 2-D tensor DMA descriptor (CDNA5 ISA) and issue
// TENSOR_LOAD_TO_LDS: tile = UCB_ROWS rows x 256 cols of f32, row stride 256.
// Must be executed by exactly one wave of the block.
// clang-23 toolchain: 6-arg builtin (g0:v4u, g1:v8i, g2:v4i, g3:v4i, v8i, cpol).
__device__ __forceinline__ void tdm_load_tile(const float* gsrc, float* lds) {
  uint64_t ga = (uint64_t)(uintptr_t)gsrc;
  uint32_t la = (uint32_t)(uintptr_t)lds;   // low 32 bits of generic ptr == LDS byte addr

  v4u g0;
  g0.x = 1u;                                           // count=1, user mode
  g0.y = la;                                           // lds_addr (bytes)
  g0.z = (uint32_t)ga;                                 // global_addr[31:0]
  g0.w = ((uint32_t)(ga >> 32) & 0x01FFFFFFu)          // global_addr[56:32]
         | (2u << 30);                                 // type=2 ("image")

  v8i g1;
  g1[0] = (int)(2u << 16);                 // workgroup_mask=0, data_size=2 (4 bytes)
  g1[1] = (int)((uint32_t)UCB_D << 16);    // tensor_dim0[15:0] in bits[63:48]
  g1[2] = (int)((uint32_t)(UCB_B * UCB_T) << 16); // dim0 hi=0; tensor_dim1[15:0] in bits[95:80]
  g1[3] = (int)((uint32_t)UCB_D << 16);    // tensor_dim1 hi=0; tile_dim0=256 in bits[127:112]
  g1[4] = UCB_ROWS;                        // tile_dim1=64, tile_dim2=0
  g1[5] = UCB_D;                           // tensor_dim0_stride[31:0] = 256 elements
  g1[6] = 0;                               // stride hi bits, tensor_dim1_stride lo = 0
  g1[7] = 0;

  v4i zz4 = {0, 0, 0, 0};                  // groups 2/3 unused (2-D tile)
  v8i zz8 = {0, 0, 0, 0, 0, 0, 0, 0};
  __builtin_amdgcn_tensor_load_to_lds(g0, g1, zz4, zz4, zz8, 0);
}
#endif

// Split tile staging: issue the DMA (wave 0), do independent work, then drain.
__device__ __forceinline__ void stage_issue(const float* __restrict__ gsrc, float* lds) {
#ifdef UCB_USE_TDM
  if (threadIdx.x < 32) tdm_load_tile(gsrc, lds);   // uniform branch: wave 0 only
#else
  for (int i = threadIdx.x; i < UCB_ROWS * UCB_D; i += 256) lds[i] = gsrc[i];
#endif
}

__device__ __forceinline__ void stage_wait() {
#ifdef UCB_USE_TDM
  if (threadIdx.x < 32) __builtin_amdgcn_s_wait_tensorcnt(0);  // issuing wave drains
#endif
  __syncthreads();
}

// Per-thread component parameters for column d: mean and 0.5/(softplus(var)+eps).
// Executed while the TDM transfer is in flight.
__device__ __forceinline__ void load_params(const float* __restrict__ mean,
                                            const float* __restrict__ variance,
                                            int d, float* m, float* iv) {
#pragma unroll
  for (int k = 0; k < UCB_K; ++k) {
    m[k] = mean[k * UCB_D + d];
    float v  = variance[k * UCB_D + d];
    float sp = __logf(1.0f + __expf(v));   // softplus
    iv[k] = 0.5f / (sp + UCB_EPS);
  }
}

// ---------------- pass 1: sum_tau partials over 64-row tiles ----------------
__global__ __launch_bounds__(256) void k_pass1(const float* __restrict__ x,
                                               const float* __restrict__ mean,
                                               const float* __restrict__ variance,
                                               float* __restrict__ p1) {
  __shared__ float tile[UCB_ROWS * UCB_D];
  const int d = threadIdx.x;
  const int blk = blockIdx.x;
  stage_issue(x + (size_t)blk * UCB_ROWS * UCB_D, tile);

  float m[UCB_K], iv[UCB_K], acc[UCB_K];
  load_params(mean, variance, d, m, iv);
#pragma unroll
  for (int k = 0; k < UCB_K; ++k) acc[k] = 0.0f;

  stage_wait();

#pragma unroll 4
  for (int r = 0; r < UCB_ROWS; ++r) {
    float xv = tile[r * UCB_D + d];
    float p[UCB_K]; float s = 0.0f;
#pragma unroll
    for (int k = 0; k < UCB_K; ++k) {
      float df = xv - m[k];
      p[k] = __expf(-df * df * iv[k]);
      s += p[k];
    }
    float inv = 1.0f / (s + UCB_EPS);
#pragma unroll
    for (int k = 0; k < UCB_K; ++k) acc[k] += p[k] * inv;
  }
#pragma unroll
  for (int k = 0; k < UCB_K; ++k)
    p1[((size_t)blk * UCB_K + k) * UCB_D + d] = acc[k];
}

// ---------------- reduce sum_tau -> 1/(sum_tau+eps) ----------------
__global__ __launch_bounds__(256) void k_reduce_sumtau(const float* __restrict__ p1,
                                                       float* __restrict__ ist) {
  const int k = blockIdx.x, d = threadIdx.x;
  float s = 0.0f;
#pragma unroll 8
  for (int b = 0; b < UCB_NB; ++b) s += p1[((size_t)b * UCB_K + k) * UCB_D + d];
  ist[k * UCB_D + d] = 1.0f / (s + UCB_EPS);
}

// ---------------- pass 2: E/V partials (per b, per t-chunk) ----------------
__global__ __launch_bounds__(256) void k_pass2(const float* __restrict__ x,
                                               const float* __restrict__ mean,
                                               const float* __restrict__ variance,
                                               const float* __restrict__ ist,
                                               float* __restrict__ p2E,
                                               float* __restrict__ p2V) {
  __shared__ float tile[UCB_ROWS * UCB_D];
  const int d = threadIdx.x;
  const int blk = blockIdx.x;                 // blk = b*8 + tchunk -> rows contiguous
  stage_issue(x + (size_t)blk * UCB_ROWS * UCB_D, tile);

  float m[UCB_K], iv[UCB_K], stk[UCB_K], sE[UCB_K], sV[UCB_K];
  load_params(mean, variance, d, m, iv);
#pragma unroll
  for (int k = 0; k < UCB_K; ++k) {
    stk[k] = ist[k * UCB_D + d];
    sE[k] = 0.0f; sV[k] = 0.0f;
  }

  stage_wait();

#pragma unroll 4
  for (int r = 0; r < UCB_ROWS; ++r) {
    float xv = tile[r * UCB_D + d];
    float p[UCB_K]; float s = 0.0f;
#pragma unroll
    for (int k = 0; k < UCB_K; ++k) {
      float df = xv - m[k];
      p[k] = __expf(-df * df * iv[k]);
      s += p[k];
    }
    float inv = 1.0f / (s + UCB_EPS);
#pragma unroll
    for (int k = 0; k < UCB_K; ++k) {
      float tau  = p[k] * inv;
      float ht   = tau * stk[k];          // hat_tau
      float prod = ht * xv;
      sE[k] += prod;
      sV[k] += ht * prod * prod;          // hat_tau * prod^2
    }
  }
#pragma unroll
  for (int k = 0; k < UCB_K; ++k) {
    size_t idx = ((size_t)blk * UCB_K + k) * UCB_D + d;
    p2E[idx] = sE[k];
    p2V[idx] = sV[k];
  }
}

// ---------------- reduce E/V over t-chunks ----------------
__global__ __launch_bounds__(256) void k_reduce_ev(const float* __restrict__ p2E,
                                                   const float* __restrict__ p2V,
                                                   float* __restrict__ Ebuf,
                                                   float* __restrict__ RSbuf) {
  const int k = blockIdx.x >> 4;          // 0..7
  const int b = blockIdx.x & 15;          // 0..15
  const int d = threadIdx.x;
  float se = 0.0f, sv = 0.0f;
#pragma unroll
  for (int tc = 0; tc < 8; ++tc) {
    size_t idx = ((size_t)(b * 8 + tc) * UCB_K + k) * UCB_D + d;
    se += p2E[idx];
    sv += p2V[idx];
  }
  const float invT = 1.0f / (float)UCB_T;
  size_t o = ((size_t)k * UCB_B + b) * UCB_D + d;
  Ebuf[o]  = se * invT;
  RSbuf[o] = rsqrtf(sv * invT + UCB_EPS);
}

// ---------------- pass 3: final output ----------------
__global__ __launch_bounds__(256) void k_pass3(const float* __restrict__ x,
                                               const float* __restrict__ mean,
                                               const float* __restrict__ variance,
                                               const float* __restrict__ Ebuf,
                                               const float* __restrict__ RSbuf,
                                               float* __restrict__ out) {
  __shared__ float tile[UCB_ROWS * UCB_D];
  const int d = threadIdx.x;
  const int blk = blockIdx.x;             // blk = b*8 + tchunk
  const int b = blk >> 3;
  stage_issue(x + (size_t)blk * UCB_ROWS * UCB_D, tile);

  float m[UCB_K], iv[UCB_K], Ek[UCB_K], rs[UCB_K];
  load_params(mean, variance, d, m, iv);
#pragma unroll
  for (int k = 0; k < UCB_K; ++k) {
    size_t o = ((size_t)k * UCB_B + b) * UCB_D + d;
    Ek[k] = Ebuf[o];
    rs[k] = RSbuf[o];
  }

  stage_wait();

  const size_t row0 = (size_t)blk * UCB_ROWS;
#pragma unroll 4
  for (int r = 0; r < UCB_ROWS; ++r) {
    float xv = tile[r * UCB_D + d];
    float p[UCB_K]; float s = 0.0f;
#pragma unroll
    for (int k = 0; k < UCB_K; ++k) {
      float df = xv - m[k];
      p[k] = __expf(-df * df * iv[k]);
      s += p[k];
    }
    float inv = 1.0f / (s + UCB_EPS);
    float o = 0.0f;
#pragma unroll
    for (int k = 0; k < UCB_K; ++k)
      o += p[k] * (xv - Ek[k]) * rs[k];
    out[(row0 + r) * UCB_D + d] = o * inv * UCB_CNORM;
  }
}

extern "C" void kernel_launch(void* const* d_in, const int* in_sizes, int n_in,
                              void* d_out, int out_size, void* d_ws, size_t ws_size,
                              hipStream_t stream) {
  (void)in_sizes; (void)n_in; (void)out_size; (void)ws_size;
  const float* x        = (const float*)d_in[0];
  const float* mean     = (const float*)d_in[1];
  const float* variance = (const float*)d_in[2];
  // d_in[3] (prior) is mathematically dead: softmax over a size-1 axis == 1.
  float* out = (float*)d_out;

  // workspace layout (floats); total ~3.26 MB
  float* p1    = (float*)d_ws;                 // [NB,K,D]  = 262144
  float* ist   = p1   + (size_t)UCB_NB * UCB_K * UCB_D;   // [K,D] = 2048
  float* p2E   = ist  + (size_t)UCB_K * UCB_D;            // [NB,K,D]
  float* p2V   = p2E  + (size_t)UCB_NB * UCB_K * UCB_D;   // [NB,K,D]
  float* Ebuf  = p2V  + (size_t)UCB_NB * UCB_K * UCB_D;   // [K,B,D] = 32768
  float* RSbuf = Ebuf + (size_t)UCB_K * UCB_B * UCB_D;    // [K,B,D]

  k_pass1<<<UCB_NB, UCB_D, 0, stream>>>(x, mean, variance, p1);
  k_reduce_sumtau<<<UCB_K, UCB_D, 0, stream>>>(p1, ist);
  k_pass2<<<UCB_NB, UCB_D, 0, stream>>>(x, mean, variance, ist, p2E, p2V);
  k_reduce_ev<<<UCB_K * UCB_B, UCB_D, 0, stream>>>(p2E, p2V, Ebuf, RSbuf);
  k_pass3<<<UCB_NB, UCB_D, 0, stream>>>(x, mean, variance, Ebuf, RSbuf, out);
}